// SolventField_25202868093415
// MI455X (gfx1250) — compile-verified
//
#include <hip/hip_runtime.h>

typedef float v2f __attribute__((ext_vector_type(2)));
typedef float v8f __attribute__((ext_vector_type(8)));

#define NRES 16384
#define TILE 16
#define NTILES (NRES / TILE)                    // 1024
#define WAVES_PER_BLOCK 8
#define JT_PER_WAVE (NTILES / WAVES_PER_BLOCK)  // 128

// ---------------------------------------------------------------------------
// Kernel 0: per-residue compare threshold thr[i] = 100 - ||x_i||^2
//           (so that  d^2 < 100  <=>  rI - 2*dot < thr[j])
//           + zero the integer density accumulators
// ---------------------------------------------------------------------------
__global__ void solvent_prep_kernel(const float* __restrict__ coords,
                                    float* __restrict__ thr,
                                    int* __restrict__ density) {
    int i = blockIdx.x * blockDim.x + threadIdx.x;
    if (i < NRES) {
        float x = coords[3 * i + 0];
        float y = coords[3 * i + 1];
        float z = coords[3 * i + 2];
        thr[i]     = 100.0f - (x * x + y * y + z * z);
        density[i] = 0;
    }
}

// ---------------------------------------------------------------------------
// Kernel 1: tiled distance test via V_WMMA_F32_16X16X4_F32.
//
// Augmented operands (K=4 slot carries the norm / affine 1):
//   A row m = [-2x_m, -2y_m, -2z_m, ||x_m||^2]
//   B col n = [ x_n,   y_n,   z_n,  1       ]
//   c[m][n] = ||x_m||^2 - 2 x_m.x_n
//   count if c[m][n] < thr[n] = 100 - ||x_n||^2   <=>   d(m,n) < 10 A
//
// f32 A (16x4) layout (wave32): lane L holds row M = L%16, K = (L/16)*2 + v.
// f32 B (4x16) mirrors it (col N = L%16, same K striping).
// C/D 16x16 f32 layout: VGPR v, lanes 0-15 -> (M=v, N=lane),
//   lanes 16-31 -> (M=v+8, N=lane-16).
// ---------------------------------------------------------------------------
__global__ __launch_bounds__(256)
void solvent_contact_kernel(const float* __restrict__ coords,
                            const float* __restrict__ thr,
                            int* __restrict__ density) {
    const int lane = threadIdx.x & 31;
    const int wave = threadIdx.x >> 5;
    const int I    = blockIdx.x * TILE;     // row tile base
    const int half = lane >> 4;             // 0: lanes 0-15, 1: lanes 16-31
    const int l16  = lane & 15;
    const int k0   = half * 2;              // K pair base: 0 or 2

    // A tile (one-time): lanes 0-15 hold (-2x, -2y), lanes 16-31 hold (-2z, rI)
    v2f a;
    {
        const int row = I + l16;
        const float x = coords[row * 3 + 0];
        const float y = coords[row * 3 + 1];
        const float z = coords[row * 3 + 2];
        const float rI = x * x + y * y + z * z;
        a.x = -2.0f * (half ? z : x);
        a.y = half ? rI : (-2.0f * y);
    }

    int cnt[8];
#pragma unroll
    for (int v = 0; v < 8; ++v) cnt[v] = 0;

    // B second-slot load offset: lanes 0-15 need y (=+1); lanes 16-31 reload z
    // (harmless, in-bounds) and then select the constant 1.0.
    const int o1 = half ? 2 : 1;

    const int jt0 = wave * JT_PER_WAVE;
    for (int jj = 0; jj < JT_PER_WAVE; ++jj) {
        const int J   = (jt0 + jj) * TILE;
        const int col = J + l16;

        v2f b;
        b.x = coords[col * 3 + k0];
        const float byl = coords[col * 3 + o1];
        b.y = half ? 1.0f : byl;                 // one v_cndmask
        const float t = thr[col];                // per-lane threshold

        v8f c = {};
        c = __builtin_amdgcn_wmma_f32_16x16x4_f32(
                /*neg_a=*/false, a, /*neg_b=*/false, b,
                /*c_mod=*/(short)0, c, /*reuse_a=*/false, /*reuse_b=*/false);

#pragma unroll
        for (int v = 0; v < 8; ++v) {
            cnt[v] += (c[v] < t) ? 1 : 0;        // v_cmp + carry add
        }
    }

    // Reduce counts across the 16 column lanes of each half (wave32 shuffles;
    // xor offsets < 16 keep the two halves independent).
#pragma unroll
    for (int v = 0; v < 8; ++v) {
        int s = cnt[v];
        s += __shfl_xor(s, 1, 32);
        s += __shfl_xor(s, 2, 32);
        s += __shfl_xor(s, 4, 32);
        s += __shfl_xor(s, 8, 32);
        if (l16 == 0) {
            // integer atomics -> order-independent, bitwise deterministic
            atomicAdd(&density[I + v + 8 * half], s);
        }
    }
}

// ---------------------------------------------------------------------------
// Kernel 2: deterministic fixed-order reduction of per-residue energies
// ---------------------------------------------------------------------------
__global__ void solvent_finalize_kernel(const int* __restrict__ density,
                                        const float* __restrict__ hydro,
                                        float* __restrict__ out) {
    __shared__ float sm[256];
    const int t = threadIdx.x;
    float local = 0.0f;
    for (int i = t; i < NRES; i += 256) {
        float dens   = (float)density[i];
        float burial = 1.0f - expf(-dens / 20.0f);
        float h      = hydro[i];
        local += (h > 0.0f) ? h * burial : h * (1.0f - burial);
    }
    sm[t] = local;
    __syncthreads();
    for (int s = 128; s > 0; s >>= 1) {
        if (t < s) sm[t] += sm[t + s];
        __syncthreads();
    }
    if (t == 0) out[0] = sm[0] / (float)NRES;
}

// ---------------------------------------------------------------------------
// Launch
// ---------------------------------------------------------------------------
extern "C" void kernel_launch(void* const* d_in, const int* in_sizes, int n_in,
                              void* d_out, int out_size, void* d_ws, size_t ws_size,
                              hipStream_t stream) {
    const float* coords = (const float*)d_in[0];   // [N, 3] f32
    const float* hydro  = (const float*)d_in[1];   // [N]    f32
    float* out          = (float*)d_out;           // scalar f32

    // workspace layout: [0, N) f32 thr ; [N, 2N) i32 density  (128 KB total)
    float* thrv  = (float*)d_ws;
    int* density = (int*)((char*)d_ws + NRES * sizeof(float));

    solvent_prep_kernel<<<NRES / 256, 256, 0, stream>>>(coords, thrv, density);
    solvent_contact_kernel<<<NTILES, 256, 0, stream>>>(coords, thrv, density);
    solvent_finalize_kernel<<<1, 256, 0, stream>>>(density, hydro, out);
}